// GNN_51539607552365
// MI455X (gfx1250) — compile-verified
//
#include <hip/hip_runtime.h>
#include <hip/hip_bf16.h>

typedef float v2f __attribute__((ext_vector_type(2)));
typedef float v8f __attribute__((ext_vector_type(8)));

#define HD 10
#define WSLOT 1552   // per-conv padded weight slot: 6*4*32*2 weights + 16 bias

__device__ __forceinline__ float actf(float v, int act) {
    if (act == 1) return fmaxf(v, 0.0f);
    if (act == 2) return 1.0f / (1.0f + __expf(-v));
    return v;
}

// ---------------------------------------------------------------------------
// Weight pre-padding: lay weights out exactly as the f32 WMMA A-matrix VGPR
// striping wants them, so the conv kernel's A load is one unconditional b64.
//   slot[(g*4+kc)*64 + lane*2 + j] = W[o, g*10 + kc*4 + lh*2 + j]  (0-padded)
//   slot[1536 + c] = bias[c] (0-padded)
// ---------------------------------------------------------------------------
struct PadJob  { const float* W; const float* b; int NG; int O; };
struct PadJobs { PadJob j[52]; };

__global__ void pad_weights_k(PadJobs jobs, float* __restrict__ out)
{
    const PadJob pj = jobs.j[blockIdx.x];
    float* dst = out + (size_t)blockIdx.x * WSLOT;
    const int nw = pj.NG * 256;
    for (int i = threadIdx.x; i < nw; i += blockDim.x) {
        const int jj = i & 1, lane = (i >> 1) & 31, chunk = i >> 6;
        const int g = chunk >> 2, kc = chunk & 3;
        const int o = lane & 15, lh = lane >> 4;
        const int kl = kc * 4 + lh * 2 + jj;
        float v = 0.0f;
        if (o < pj.O && kl < HD) v = pj.W[o * (pj.NG * HD) + g * HD + kl];
        dst[i] = v;
    }
    for (int c = threadIdx.x; c < 16; c += blockDim.x)
        dst[1536 + c] = (c < pj.O) ? pj.b[c] : 0.0f;
}

// ---------------------------------------------------------------------------
// Fused 1x1-conv GEMM on WMMA f32 16x16x4.
//   Sources are NHWC-16 ("C16"): [pix][16], first HD channels real.
//   A (SRC0) = padded weights (from slot), B (SRC1) = activations.
//   K = NG groups of 16 padded channels; each wave owns 4 D tiles = 64 pixels.
//   D tile: lane L holds pixel n=L%16, out-channels m = 8*(L/16)+r in v[r].
// ---------------------------------------------------------------------------
template<int NG>
__global__ __launch_bounds__(128) void conv1x1_k(
    const float* __restrict__ s0, const float* __restrict__ s1,
    const float* __restrict__ s2, const float* __restrict__ s3,
    const float* __restrict__ s4, const float* __restrict__ s5,
    const float* __restrict__ Wp, int O, int act, int dense,
    float* __restrict__ dst)
{
    const float* S[6] = {s0, s1, s2, s3, s4, s5};
    const int lane = threadIdx.x & 31;
    const int wave = threadIdx.x >> 5;
    const int lm = lane & 15;
    const int lh = lane >> 4;
    const size_t pixBase = ((size_t)blockIdx.x * 4 + wave) * 64;

    v8f acc[4];
#pragma unroll
    for (int t = 0; t < 4; ++t)
#pragma unroll
        for (int r = 0; r < 8; ++r) acc[t][r] = 0.0f;

#pragma unroll
    for (int g = 0; g < NG; ++g) {
        const float* __restrict__ sg = S[g];
#pragma unroll
        for (int kc = 0; kc < 4; ++kc) {
            const int klo = kc * 4 + lh * 2;
            const v2f a = *(const v2f*)(Wp + ((g * 4 + kc) << 6) + (lane << 1));
#pragma unroll
            for (int t = 0; t < 4; ++t) {
                const float* bp = sg + (pixBase + (size_t)t * 16 + lm) * 16 + klo;
                const v2f b = *(const v2f*)bp;
                acc[t] = __builtin_amdgcn_wmma_f32_16x16x4_f32(
                    false, a, false, b, (short)0, acc[t], false, false);
            }
        }
    }

    const float* bias16 = Wp + 1536;
#pragma unroll
    for (int t = 0; t < 4; ++t) {
        const size_t pix = pixBase + (size_t)t * 16 + lm;
        if (dense) {
            if (lh == 0) dst[pix] = actf(acc[t][0] + bias16[0], act);
        } else {
            float* dp = dst + pix * 16 + lh * 8;
#pragma unroll
            for (int r = 0; r < 8; ++r) {
                const int m = lh * 8 + r;
                const float v = actf(acc[t][r] + bias16[m], act);
                dp[r] = (m < O) ? v : 0.0f;
            }
        }
    }
}

// ------------------------- layout / elementwise kernels --------------------

// NCHW (C=HD) -> C16 NHWC, zero padded.  Block: 16 pixels x 16 channels.
__global__ void pack_c16_k(const float* __restrict__ in, float* __restrict__ out,
                           int HW, int npix)
{
    const int p = threadIdx.x >> 4;
    const int c = threadIdx.x & 15;
    const int pix = blockIdx.x * 16 + p;
    if (pix >= npix) return;
    const int b = pix / HW, hw = pix % HW;
    float v = 0.0f;
    if (c < HD) v = in[((size_t)(b * HD + c)) * HW + hw];
    out[(size_t)pix * 16 + c] = v;
}

// C16 NHWC -> NCHW (C=HD)
__global__ void c16_to_nchw_k(const float* __restrict__ in, float* __restrict__ out,
                              int HW, int npix)
{
    const int c = threadIdx.x >> 4;
    const int p = threadIdx.x & 15;
    const int pix = blockIdx.x * 16 + p;
    if (pix >= npix || c >= HD) return;
    const int b = pix / HW, hw = pix % HW;
    out[((size_t)(b * HD + c)) * HW + hw] = in[(size_t)pix * 16 + c];
}

// grouped attention: out[g*npix+pix] = sigmoid(dot(W[g,:], src_g[pix,:]) + b[g])
__global__ void grouped_att_k(const float* s0, const float* s1, const float* s2,
                              const float* s3, const float* s4, const float* s5,
                              const float* __restrict__ W, const float* __restrict__ b,
                              int G, int npix, float* __restrict__ dst)
{
    const int i = blockIdx.x * blockDim.x + threadIdx.x;
    if (i >= G * npix) return;
    const int g = i / npix, pix = i - g * npix;
    const float* S[6] = {s0, s1, s2, s3, s4, s5};
    const float* x = S[g] + (size_t)pix * 16;
    float acc = b[g];
#pragma unroll
    for (int c = 0; c < HD; ++c) acc += W[g * HD + c] * x[c];
    dst[i] = 1.0f / (1.0f + __expf(-acc));
}

// align-corners bilinear on C16 tensors
__global__ void bilinear_c16_k(const float* __restrict__ in, float* __restrict__ out,
                               int Hin, int Win, int Hout, int Wout, int Bn)
{
    const int i = blockIdx.x * blockDim.x + threadIdx.x;
    const int total = Bn * Hout * Wout;
    if (i >= total) return;
    const int ox = i % Wout; int t = i / Wout;
    const int oy = t % Hout; const int b = t / Hout;
    const float fy = (float)oy * (float)(Hin - 1) / (float)(Hout - 1);
    const float fx = (float)ox * (float)(Win - 1) / (float)(Wout - 1);
    int y0 = (int)fy; if (y0 > Hin - 1) y0 = Hin - 1;
    int x0 = (int)fx; if (x0 > Win - 1) x0 = Win - 1;
    const int y1 = min(y0 + 1, Hin - 1), x1 = min(x0 + 1, Win - 1);
    const float wy = fy - (float)y0, wx = fx - (float)x0;
    const float* p00 = in + (((size_t)b * Hin + y0) * Win + x0) * 16;
    const float* p01 = in + (((size_t)b * Hin + y0) * Win + x1) * 16;
    const float* p10 = in + (((size_t)b * Hin + y1) * Win + x0) * 16;
    const float* p11 = in + (((size_t)b * Hin + y1) * Win + x1) * 16;
    float* op = out + (size_t)i * 16;
#pragma unroll
    for (int c = 0; c < 16; ++c) {
        const float top = p00[c] * (1.0f - wx) + p01[c] * wx;
        const float bot = p10[c] * (1.0f - wx) + p11[c] * wx;
        op[c] = top * (1.0f - wy) + bot * wy;
    }
}

// align-corners bilinear on a dense map, writing into NCHW d_out plane
__global__ void bilin_dense_out_k(const float* __restrict__ in, float* __restrict__ out,
                                  int Hin, int Win, int Hout, int Wout, int Bn,
                                  size_t bstride)
{
    const int i = blockIdx.x * blockDim.x + threadIdx.x;
    const int total = Bn * Hout * Wout;
    if (i >= total) return;
    const int ox = i % Wout; int t = i / Wout;
    const int oy = t % Hout; const int b = t / Hout;
    const float fy = (float)oy * (float)(Hin - 1) / (float)(Hout - 1);
    const float fx = (float)ox * (float)(Win - 1) / (float)(Wout - 1);
    int y0 = (int)fy; if (y0 > Hin - 1) y0 = Hin - 1;
    int x0 = (int)fx; if (x0 > Win - 1) x0 = Win - 1;
    const int y1 = min(y0 + 1, Hin - 1), x1 = min(x0 + 1, Win - 1);
    const float wy = fy - (float)y0, wx = fx - (float)x0;
    const float* pb = in + (size_t)b * Hin * Win;
    const float top = pb[y0 * Win + x0] * (1.0f - wx) + pb[y0 * Win + x1] * wx;
    const float bot = pb[y1 * Win + x0] * (1.0f - wx) + pb[y1 * Win + x1] * wx;
    out[(size_t)b * bstride + (size_t)oy * Wout + ox] = top * (1.0f - wy) + bot * wy;
}

__global__ void nearest_c16_k(const float* __restrict__ in, float* __restrict__ out,
                              int Hin, int Win, int Hout, int Wout, int Bn)
{
    const int i = blockIdx.x * blockDim.x + threadIdx.x;
    const int total = Bn * Hout * Wout * 16;
    if (i >= total) return;
    const int c = i & 15; const int opix = i >> 4;
    const int ox = opix % Wout; int t = opix / Wout;
    const int oy = t % Hout; const int b = t / Hout;
    const int yi = oy * Hin / Hout, xi = ox * Win / Wout;
    out[i] = in[(((size_t)b * Hin + yi) * Win + xi) * 16 + c];
}

// out = src_c16 * dense_map (broadcast over 16 channels)
__global__ void mul_bcast_k(const float* __restrict__ s, const float* __restrict__ dmap,
                            float* __restrict__ o, int n)
{
    const int i = blockIdx.x * blockDim.x + threadIdx.x;
    if (i < n) o[i] = s[i] * dmap[i >> 4];
}

// o (+)= pd * (2 - dense_map); init==1 overwrites
__global__ void pd_acc_k(const float* __restrict__ pd, const float* __restrict__ dmap,
                         float* __restrict__ o, int n, int init)
{
    const int i = blockIdx.x * blockDim.x + threadIdx.x;
    if (i >= n) return;
    const float v = pd[i] * (2.0f - dmap[i >> 4]);
    o[i] = init ? v : (o[i] + v);
}

__global__ void avg2_k(const float* __restrict__ a, const float* __restrict__ b,
                       float* __restrict__ o, int n)
{
    const int i = blockIdx.x * blockDim.x + threadIdx.x;
    if (i < n) o[i] = 0.5f * (a[i] + b[i]);
}

// att channels 2..7: (att_fp + att_hp + dp_att_p) / 3 -> NCHW d_out
__global__ void attp_out_k(const float* __restrict__ attf, const float* __restrict__ atth,
                           const float* __restrict__ dattp, float* __restrict__ out,
                           int npix, int HW)
{
    const int i = blockIdx.x * blockDim.x + threadIdx.x;
    if (i >= 6 * npix) return;
    const int ch = i / npix, pix = i - ch * npix;
    const int b = pix / HW, hw = pix % HW;
    const float v = (attf[(size_t)ch * npix + pix] + atth[(size_t)ch * npix + pix] +
                     dattp[(size_t)ch * npix + pix]) * (1.0f / 3.0f);
    out[((size_t)(b * 8 + 2 + ch)) * HW + hw] = v;
}

// ---------------------------------------------------------------------------

extern "C" void kernel_launch(void* const* d_in, const int* in_sizes, int n_in,
                              void* d_out, int out_size, void* d_ws, size_t ws_size,
                              hipStream_t stream)
{
    (void)in_sizes; (void)n_in; (void)out_size; (void)ws_size;
#define IN(i) ((const float*)d_in[(i)])

    const int Bn = 32;
    const int HWP = 128 * 128, HWF = 64 * 64;
    const int NPIXP = Bn * HWP;           // 524288
    const int NPIXF = Bn * HWF;           // 131072
    const size_t C16P = (size_t)NPIXP * 16;
    const size_t C16F = (size_t)NPIXF * 16;
    const size_t DENP = (size_t)NPIXP;
    const size_t DENF = (size_t)NPIXF;

    float* ws = (float*)d_ws;
    size_t off = 0;
    auto alloc = [&](size_t n) { size_t r = off; off += n; return r; };
    const size_t XPC    = alloc(6 * C16P);
    const size_t XHC    = alloc(2 * C16F);
    const size_t XFC    = alloc(C16F);
    const size_t TMPP   = alloc(C16P);
    const size_t F_COMPH = alloc(C16F);
    const size_t F_COMPP = alloc(C16F);
    const size_t F_HID   = alloc(C16F);
    const size_t F_PD    = alloc(C16F);
    const size_t F_CMP   = alloc(C16F);
    const size_t F_DP    = alloc(C16F);
    const size_t F_AXF   = alloc(C16F);
    const size_t F_OUT   = alloc(C16F);
    const size_t DATTF  = alloc(2 * DENF);
    const size_t ATTUF  = alloc(DENF);
    const size_t ATTLF  = alloc(DENF);
    const size_t ATTH0  = alloc(DENF);
    const size_t ATTH1  = alloc(DENF);
    const size_t P_XFP  = alloc(C16P);
    const size_t P_XH0  = alloc(C16P);
    const size_t P_XH1  = alloc(C16P);
    const size_t P_DECF = alloc(C16P);
    const size_t P_DECH = alloc(C16P);
    const size_t P_ACC  = alloc(C16P);
    const size_t P_PD   = alloc(C16P);
    const size_t P_HID  = alloc(C16P);
    const size_t P_OUT  = alloc(C16P);
    const size_t DATTP  = alloc(6 * DENP);
    const size_t ATTFP  = alloc(6 * DENP);
    const size_t ATTHP  = alloc(6 * DENP);
    const size_t PADW   = alloc(52 * WSLOT);

    float* const xpc[6] = {ws + XPC + 0 * C16P, ws + XPC + 1 * C16P, ws + XPC + 2 * C16P,
                           ws + XPC + 3 * C16P, ws + XPC + 4 * C16P, ws + XPC + 5 * C16P};
    float* const xhc0 = ws + XHC;
    float* const xhc1 = ws + XHC + C16F;
    float* const xfc  = ws + XFC;

    float* out_xp  = (float*)d_out;
    float* out_xh  = out_xp + (size_t)6 * Bn * HD * HWP;
    float* out_xf  = out_xh + (size_t)2 * Bn * HD * HWF;
    float* out_att = out_xf + (size_t)Bn * HD * HWF;

    // ---------------- register + pre-pad all conv weights ----------------
    PadJobs jobs{};
    int ns = 0;
    auto reg = [&](const float* W, const float* b, int NG, int O) -> float* {
        jobs.j[ns].W = W; jobs.j[ns].b = b; jobs.j[ns].NG = NG; jobs.j[ns].O = O;
        float* p = ws + PADW + (size_t)ns * WSLOT; ++ns; return p;
    };
    float* Pw_comph = reg(IN(3),  IN(4),  2, HD);
    float* Pw_compp = reg(IN(5),  IN(6),  6, HD);
    float* Pw_fupd  = reg(IN(7),  IN(8),  3, HD);
    float* Pw_hpd1  = reg(IN(21), IN(22), 2, HD);
    float* Pw_hpd2  = reg(IN(23), IN(24), 1, HD);
    float* Pw_cphu  = reg(IN(17), IN(18), 4, HD);
    float* Pw_fhu1  = reg(IN(9),  IN(10), 2, HD);
    float* Pw_fhu2  = reg(IN(11), IN(12), 1, 1);
    float* Pw_updu  = reg(IN(27), IN(28), 4, HD);
    float* Pw_cphl  = reg(IN(19), IN(20), 2, HD);
    float* Pw_fhl1  = reg(IN(13), IN(14), 2, HD);
    float* Pw_fhl2  = reg(IN(15), IN(16), 1, 1);
    float *Pw_dfp1[6], *Pw_dfp2[6], *Pw_dhp1[6], *Pw_dhp2[6], *Pw_upd[6];
    for (int j = 0; j < 6; ++j) {
        Pw_dfp1[j] = reg(IN(35) + j * 200, IN(36) + j * HD, 2, HD);
        Pw_dfp2[j] = reg(IN(37) + j * HD,  IN(38) + j,      1, 1);
        Pw_dhp1[j] = reg(IN(39) + j * 200, IN(40) + j * HD, 2, HD);
        Pw_dhp2[j] = reg(IN(41) + j * HD,  IN(42) + j,      1, 1);
        Pw_upd[j]  = reg(IN(43) + j * 400, IN(44) + j * HD, 4, HD);
    }
    float *Pw_e1[5], *Pw_e2[5];
    for (int e = 0; e < 5; ++e) {
        Pw_e1[e] = reg(IN(29) + e * 200, IN(30) + e * HD, 2, HD);
        Pw_e2[e] = reg(IN(31) + e * 100, IN(32) + e * HD, 1, HD);
    }
    pad_weights_k<<<ns, 64, 0, stream>>>(jobs, ws + PADW);

    auto CONV = [&](int NG, const float* s0, const float* s1, const float* s2,
                    const float* s3, const float* s4, const float* s5,
                    const float* Wp, int O, int npix, int act, int dense, float* dst) {
        dim3 g(npix / 256), blk(128);
        switch (NG) {
        case 1: conv1x1_k<1><<<g, blk, 0, stream>>>(s0,s1,s2,s3,s4,s5,Wp,O,act,dense,dst); break;
        case 2: conv1x1_k<2><<<g, blk, 0, stream>>>(s0,s1,s2,s3,s4,s5,Wp,O,act,dense,dst); break;
        case 3: conv1x1_k<3><<<g, blk, 0, stream>>>(s0,s1,s2,s3,s4,s5,Wp,O,act,dense,dst); break;
        case 4: conv1x1_k<4><<<g, blk, 0, stream>>>(s0,s1,s2,s3,s4,s5,Wp,O,act,dense,dst); break;
        default: conv1x1_k<6><<<g, blk, 0, stream>>>(s0,s1,s2,s3,s4,s5,Wp,O,act,dense,dst); break;
        }
    };
    auto EW = [&](int n) { return dim3((n + 255) / 256); };

    // ---------------- pack inputs to C16 ----------------
    for (int j = 0; j < 6; ++j)
        pack_c16_k<<<NPIXP / 16, 256, 0, stream>>>(IN(0) + (size_t)j * Bn * HD * HWP,
                                                   xpc[j], HWP, NPIXP);
    pack_c16_k<<<NPIXF / 16, 256, 0, stream>>>(IN(1), xhc0, HWF, NPIXF);
    pack_c16_k<<<NPIXF / 16, 256, 0, stream>>>(IN(1) + (size_t)Bn * HD * HWF, xhc1, HWF, NPIXF);
    pack_c16_k<<<NPIXF / 16, 256, 0, stream>>>(IN(2), xfc, HWF, NPIXF);

    // ---------------- full level ----------------
    CONV(2, xhc0, xhc1, 0, 0, 0, 0, Pw_comph, HD, NPIXF, 1, 0, ws + F_COMPH);
    CONV(6, xpc[0], xpc[1], xpc[2], xpc[3], xpc[4], xpc[5], Pw_compp, HD, NPIXP, 1, 0, ws + TMPP);
    bilinear_c16_k<<<EW(NPIXF), 256, 0, stream>>>(ws + TMPP, ws + F_COMPP, 128, 128, 64, 64, Bn);
    CONV(3, xfc, ws + F_COMPH, ws + F_COMPP, 0, 0, 0, Pw_fupd, HD, NPIXF, 1, 0, ws + F_OUT);
    c16_to_nchw_k<<<NPIXF / 16, 256, 0, stream>>>(ws + F_OUT, out_xf, HWF, NPIXF);

    // ---------------- half level ----------------
    grouped_att_k<<<EW(2 * NPIXF), 256, 0, stream>>>(xhc0, xhc1, 0, 0, 0, 0,
                                                     IN(25), IN(26), 2, NPIXF, ws + DATTF);
    CONV(2, xhc0, xhc1, 0, 0, 0, 0, Pw_hpd1, HD, NPIXF, 1, 0, ws + F_HID);
    CONV(1, ws + F_HID, 0, 0, 0, 0, 0, Pw_hpd2, HD, NPIXF, 1, 0, ws + F_PD);

    // upper half
    CONV(4, xpc[0], xpc[1], xpc[2], xpc[3], 0, 0, Pw_cphu, HD, NPIXP, 1, 0, ws + TMPP);
    bilinear_c16_k<<<EW(NPIXF), 256, 0, stream>>>(ws + TMPP, ws + F_CMP, 128, 128, 64, 64, Bn);
    pd_acc_k<<<EW(16 * NPIXF), 256, 0, stream>>>(ws + F_PD, ws + DATTF + DENF, ws + F_DP,
                                                 16 * NPIXF, 1);
    CONV(2, xfc, xhc0, 0, 0, 0, 0, Pw_fhu1, HD, NPIXF, 1, 0, ws + F_HID);
    CONV(1, ws + F_HID, 0, 0, 0, 0, 0, Pw_fhu2, 1, NPIXF, 2, 1, ws + ATTUF);
    mul_bcast_k<<<EW(16 * NPIXF), 256, 0, stream>>>(xfc, ws + ATTUF, ws + F_AXF, 16 * NPIXF);
    CONV(4, xhc0, ws + F_CMP, ws + F_AXF, ws + F_DP, 0, 0, Pw_updu, HD, NPIXF, 1, 0, ws + F_OUT);
    c16_to_nchw_k<<<NPIXF / 16, 256, 0, stream>>>(ws + F_OUT, out_xh, HWF, NPIXF);

    // lower half
    CONV(2, xpc[4], xpc[5], 0, 0, 0, 0, Pw_cphl, HD, NPIXP, 1, 0, ws + TMPP);
    bilinear_c16_k<<<EW(NPIXF), 256, 0, stream>>>(ws + TMPP, ws + F_CMP, 128, 128, 64, 64, Bn);
    pd_acc_k<<<EW(16 * NPIXF), 256, 0, stream>>>(ws + F_PD, ws + DATTF, ws + F_DP, 16 * NPIXF, 1);
    CONV(2, xfc, xhc1, 0, 0, 0, 0, Pw_fhl1, HD, NPIXF, 1, 0, ws + F_HID);
    CONV(1, ws + F_HID, 0, 0, 0, 0, 0, Pw_fhl2, 1, NPIXF, 2, 1, ws + ATTLF);
    mul_bcast_k<<<EW(16 * NPIXF), 256, 0, stream>>>(xfc, ws + ATTLF, ws + F_AXF, 16 * NPIXF);
    CONV(4, xhc1, ws + F_CMP, ws + F_AXF, ws + F_DP, 0, 0, Pw_updu, HD, NPIXF, 1, 0, ws + F_OUT);
    c16_to_nchw_k<<<NPIXF / 16, 256, 0, stream>>>(ws + F_OUT, out_xh + (size_t)Bn * HD * HWF,
                                                  HWF, NPIXF);

    // att_h -> att channels 0,1 (bilinear to part resolution, straight to d_out)
    avg2_k<<<EW(NPIXF), 256, 0, stream>>>(ws + ATTUF, ws + DATTF, ws + ATTH0, NPIXF);
    avg2_k<<<EW(NPIXF), 256, 0, stream>>>(ws + ATTLF, ws + DATTF + DENF, ws + ATTH1, NPIXF);
    bilin_dense_out_k<<<EW(NPIXP), 256, 0, stream>>>(ws + ATTH0, out_att, 64, 64, 128, 128,
                                                     Bn, (size_t)8 * HWP);
    bilin_dense_out_k<<<EW(NPIXP), 256, 0, stream>>>(ws + ATTH1, out_att + HWP, 64, 64, 128, 128,
                                                     Bn, (size_t)8 * HWP);

    // ---------------- part level ----------------
    nearest_c16_k<<<EW(16 * NPIXP), 256, 0, stream>>>(xfc, ws + P_XFP, 64, 64, 128, 128, Bn);
    bilinear_c16_k<<<EW(NPIXP), 256, 0, stream>>>(xhc0, ws + P_XH0, 64, 64, 128, 128, Bn);
    bilinear_c16_k<<<EW(NPIXP), 256, 0, stream>>>(xhc1, ws + P_XH1, 64, 64, 128, 128, Bn);
    grouped_att_k<<<EW(6 * NPIXP), 256, 0, stream>>>(xpc[0], xpc[1], xpc[2], xpc[3], xpc[4],
                                                     xpc[5], IN(33), IN(34), 6, NPIXP, ws + DATTP);

    static const int EA[5] = {0, 1, 2, 1, 4};
    static const int EB[5] = {1, 2, 3, 4, 5};
    // per part: {edge index, other endpoint}
    static const int EJ0[][2] = {{0, 1}};
    static const int EJ1[][2] = {{0, 0}, {1, 2}, {3, 4}};
    static const int EJ2[][2] = {{1, 1}, {2, 3}};
    static const int EJ3[][2] = {{2, 2}};
    static const int EJ4[][2] = {{3, 1}, {4, 5}};
    static const int EJ5[][2] = {{4, 4}};
    const int (*EJ[6])[2] = {EJ0, EJ1, EJ2, EJ3, EJ4, EJ5};
    const int NEJ[6] = {1, 3, 2, 1, 2, 1};

    for (int j = 0; j < 6; ++j) {
        // decomposition full->part
        CONV(2, ws + P_XFP, xpc[j], 0, 0, 0, 0, Pw_dfp1[j], HD, NPIXP, 1, 0, ws + P_HID);
        CONV(1, ws + P_HID, 0, 0, 0, 0, 0, Pw_dfp2[j], 1, NPIXP, 2, 1,
             ws + ATTFP + (size_t)j * DENP);
        mul_bcast_k<<<EW(16 * NPIXP), 256, 0, stream>>>(ws + P_XFP,
            ws + ATTFP + (size_t)j * DENP, ws + P_DECF, 16 * NPIXP);
        // decomposition half->part
        const float* xs = (j < 4) ? (ws + P_XH0) : (ws + P_XH1);
        CONV(2, xs, xpc[j], 0, 0, 0, 0, Pw_dhp1[j], HD, NPIXP, 1, 0, ws + P_HID);
        CONV(1, ws + P_HID, 0, 0, 0, 0, 0, Pw_dhp2[j], 1, NPIXP, 2, 1,
             ws + ATTHP + (size_t)j * DENP);
        mul_bcast_k<<<EW(16 * NPIXP), 256, 0, stream>>>(xs,
            ws + ATTHP + (size_t)j * DENP, ws + P_DECH, 16 * NPIXP);
        // part-dependency accumulation over incident edges
        for (int k = 0; k < NEJ[j]; ++k) {
            const int e = EJ[j][k][0], other = EJ[j][k][1];
            CONV(2, xpc[EA[e]], xpc[EB[e]], 0, 0, 0, 0, Pw_e1[e], HD, NPIXP, 1, 0, ws + P_HID);
            CONV(1, ws + P_HID, 0, 0, 0, 0, 0, Pw_e2[e], HD, NPIXP, 1, 0, ws + P_PD);
            pd_acc_k<<<EW(16 * NPIXP), 256, 0, stream>>>(ws + P_PD,
                ws + DATTP + (size_t)other * DENP, ws + P_ACC, 16 * NPIXP, k == 0 ? 1 : 0);
        }
        // part update
        CONV(4, xpc[j], ws + P_DECF, ws + P_DECH, ws + P_ACC, 0, 0,
             Pw_upd[j], HD, NPIXP, 1, 0, ws + P_OUT);
        c16_to_nchw_k<<<NPIXP / 16, 256, 0, stream>>>(ws + P_OUT,
            out_xp + (size_t)j * Bn * HD * HWP, HWP, NPIXP);
    }

    // att channels 2..7
    attp_out_k<<<EW(6 * NPIXP), 256, 0, stream>>>(ws + ATTFP, ws + ATTHP, ws + DATTP,
                                                  out_att, NPIXP, HWP);
#undef IN
}